// QuadraticAttention_7129645711395
// MI455X (gfx1250) — compile-verified
//
#include <hip/hip_runtime.h>
#include <cstddef>

typedef float v2f __attribute__((ext_vector_type(2)));
typedef float v8f __attribute__((ext_vector_type(8)));

#define D_MODEL 768
#define N_HEAD  12
#define D_HEAD  64
#define N_CTX   2048
#define BATCH   2
#define M_ROWS  (BATCH * N_CTX)   // 4096

// D = A(16x4) * B(4x16) + C, full fp32 (V_WMMA_F32_16X16X4_F32)
__device__ __forceinline__ v8f wmma_f32(v2f a, v2f b, v8f c) {
  return __builtin_amdgcn_wmma_f32_16x16x4_f32(false, a, false, b, (short)0, c,
                                               false, false);
}

// 16(M) x 64(N) x 768(K) strip per wave: 1 A load amortized over 4 WMMAs.
// All row pointers already include the per-lane +2*hi K offset.
// A: lane(lo) = row m0+lo, VGPR0/1 = K k+2*hi, k+2*hi+1  (ISA 7.12.2 16x4 f32)
// B: lane(lo) = col n+lo,  VGPR0/1 = K k+2*hi, k+2*hi+1
__device__ __forceinline__ void gemm_strip_k768(const float* __restrict__ xrow,
                                                const float* __restrict__ w0,
                                                const float* __restrict__ w1,
                                                const float* __restrict__ w2,
                                                const float* __restrict__ w3,
                                                v8f acc[4]) {
#pragma unroll 4
  for (int k = 0; k < D_MODEL; k += 4) {
    v2f a  = *(const v2f*)(xrow + k);
    v2f b0 = *(const v2f*)(w0 + k);
    v2f b1 = *(const v2f*)(w1 + k);
    v2f b2 = *(const v2f*)(w2 + k);
    v2f b3 = *(const v2f*)(w3 + k);
    acc[0] = wmma_f32(a, b0, acc[0]);
    acc[1] = wmma_f32(a, b1, acc[1]);
    acc[2] = wmma_f32(a, b2, acc[2]);
    acc[3] = wmma_f32(a, b3, acc[3]);
  }
}

// C[m,n] = sum_k X[m,k]*W[n,k] + bias[n].
// Each wave's 64 N-columns are exactly one head (n0 is a multiple of 64).
// vmode==0: store [B, H, S, D_HEAD]   (Q, K)
// vmode==1: store [B, H, D_HEAD, S]   (V transposed, for contiguous PV loads)
__global__ __launch_bounds__(256) void qkv_gemm_kernel(
    const float* __restrict__ X, const float* __restrict__ W,
    const float* __restrict__ bias, float* __restrict__ out, int vmode) {
  const int wave = threadIdx.x >> 5;
  const int lane = threadIdx.x & 31;
  const int lo = lane & 15, hi = lane >> 4;
  const int m0 = blockIdx.x * 32 + (wave >> 2) * 16;
  const int n0 = blockIdx.y * 256 + (wave & 3) * 64;
  const float* xrow  = X + (size_t)(m0 + lo) * D_MODEL + 2 * hi;
  const float* wbase = W + (size_t)(n0 + lo) * D_MODEL + 2 * hi;
  v8f acc[4] = {{}, {}, {}, {}};
  gemm_strip_k768(xrow, wbase, wbase + 16 * D_MODEL, wbase + 32 * D_MODEL,
                  wbase + 48 * D_MODEL, acc);
  const int h = n0 >> 6;
#pragma unroll
  for (int j = 0; j < 4; ++j) {
    const int d = j * 16 + lo;
    const float bv = bias[n0 + d];
#pragma unroll
    for (int r = 0; r < 8; ++r) {
      const int m = m0 + r + 8 * hi;      // C layout: lanes16-31 hold M+8
      const int b = m >> 11, s = m & (N_CTX - 1);
      const size_t hb = (size_t)b * N_HEAD + h;
      if (vmode)
        out[(hb * D_HEAD + d) * N_CTX + s] = acc[j][r] + bv;
      else
        out[(hb * N_CTX + s) * D_HEAD + d] = acc[j][r] + bv;
    }
  }
}

// out[m,n] = sum_k X[m,k]*W[n,k]   (final projection, flat [4096,768] store)
__global__ __launch_bounds__(256) void out_gemm_kernel(
    const float* __restrict__ X, const float* __restrict__ W,
    float* __restrict__ out) {
  const int wave = threadIdx.x >> 5;
  const int lane = threadIdx.x & 31;
  const int lo = lane & 15, hi = lane >> 4;
  const int m0 = blockIdx.x * 32 + (wave >> 2) * 16;
  const int n0 = blockIdx.y * 256 + (wave & 3) * 64;
  const float* xrow  = X + (size_t)(m0 + lo) * D_MODEL + 2 * hi;
  const float* wbase = W + (size_t)(n0 + lo) * D_MODEL + 2 * hi;
  v8f acc[4] = {{}, {}, {}, {}};
  gemm_strip_k768(xrow, wbase, wbase + 16 * D_MODEL, wbase + 32 * D_MODEL,
                  wbase + 48 * D_MODEL, acc);
#pragma unroll
  for (int j = 0; j < 4; ++j) {
#pragma unroll
    for (int r = 0; r < 8; ++r) {
      const int m = m0 + r + 8 * hi;
      out[(size_t)m * D_MODEL + n0 + j * 16 + lo] = acc[j][r];
    }
  }
}

// RMSNorm(g) then RoPE, in place on [B*H*S, 64] rows. One wave per row;
// lane i owns the rotation pair (i, i+32). wave32 shuffle reduction.
__global__ __launch_bounds__(256) void norm_rope_kernel(
    float* __restrict__ qk, const float* __restrict__ g) {
  const int lane = threadIdx.x & 31;
  const int row = blockIdx.x * 8 + (threadIdx.x >> 5);
  const int s = row & (N_CTX - 1);          // layout [B,H,S]: s is fastest
  float* p = qk + (size_t)row * D_HEAD;
  const float x1 = p[lane];
  const float x2 = p[lane + 32];
  float ss = x1 * x1 + x2 * x2;
#pragma unroll
  for (int off = 16; off > 0; off >>= 1) ss += __shfl_xor(ss, off, 32);
  const float inv = rsqrtf(ss * (1.0f / 64.0f) + 1e-6f);
  const float xn1 = x1 * inv * g[lane];
  const float xn2 = x2 * inv * g[lane + 32];
  // inv_freq[i] = ROPE_BASE^(-i/32) = 2^(-11*i/32)
  const float freq = (float)s * exp2f((float)lane * (-11.0f / 32.0f));
  const float c = __cosf(freq), sn = __sinf(freq);
  p[lane]      = xn1 * c + xn2 * sn;
  p[lane + 32] = xn2 * c - xn1 * sn;
}

// Fused causal quadratic attention: Z[16x64] tile per wave.
// pattern = (Q.Kt/64)^2 * tril, Z = pattern @ V. No softmax -> no carry state.
// Q, K: [B,H,S,64]; V: [B,H,64,S] (transposed so PV B-loads are contiguous b64).
__global__ __launch_bounds__(256) void attn_kernel(
    const float* __restrict__ Q, const float* __restrict__ K,
    const float* __restrict__ V, float* __restrict__ Z) {
  __shared__ float lds[8 * 256];            // per-wave 16x16 P staging
  const int wave = threadIdx.x >> 5;
  const int lane = threadIdx.x & 31;
  const int lo = lane & 15, hi = lane >> 4;
  const int wid = blockIdx.x * 8 + wave;    // [0, B*H*128)
  const int head = wid >> 7;                // 128 q-tiles per head
  const int qt = wid & 127;
  const float* Qh = Q + (size_t)head * N_CTX * D_HEAD;
  const float* Kh = K + (size_t)head * N_CTX * D_HEAD;
  const float* Vh = V + (size_t)head * D_HEAD * N_CTX;   // [D,S]
  float* pl = lds + wave * 256;

  // Preload Q tile in A-matrix layout for all 16 K-steps (d = 0..63)
  v2f qa[16];
  {
    const float* qrow = Qh + (size_t)(qt * 16 + lo) * D_HEAD + 2 * hi;
#pragma unroll
    for (int kk = 0; kk < 16; ++kk) qa[kk] = *(const v2f*)(qrow + kk * 4);
  }

  v8f z0 = {}, z1 = {}, z2 = {}, z3 = {};
  for (int kt = 0; kt <= qt; ++kt) {
    const float* krow = Kh + (size_t)(kt * 16 + lo) * D_HEAD + 2 * hi;
    __builtin_prefetch(krow + 16 * D_HEAD, 0, 0);   // next K tile -> global_prefetch
    // S = Q x K^T : 16 WMMA steps over d
    v8f sc = {};
#pragma unroll
    for (int kk = 0; kk < 16; ++kk) {
      v2f kb = *(const v2f*)(krow + kk * 4);
      sc = wmma_f32(qa[kk], kb, sc);
    }
    // pattern = (S/64)^2, causal mask on diagonal tile; stage to LDS
    // (same-wave DS ops are hardware in-order: store->load, no block barrier)
#pragma unroll
    for (int r = 0; r < 8; ++r) {
      float pv = sc[r] * (1.0f / 64.0f);
      pv = pv * pv;
      if (kt == qt && lo > r + 8 * hi) pv = 0.0f;   // col N > row M => masked
      pl[(r + 8 * hi) * 16 + lo] = pv;
    }
    // Z += P x V : P re-read in A layout; V^T gives contiguous (s,s+1) pairs.
#pragma unroll
    for (int kk2 = 0; kk2 < 4; ++kk2) {
      v2f pa = *(const v2f*)(pl + lo * 16 + kk2 * 4 + 2 * hi);
      const float* vcol = Vh + (size_t)lo * N_CTX + kt * 16 + kk2 * 4 + 2 * hi;
      v2f v0 = *(const v2f*)(vcol);
      v2f v1 = *(const v2f*)(vcol + 16 * N_CTX);
      v2f v2 = *(const v2f*)(vcol + 32 * N_CTX);
      v2f v3 = *(const v2f*)(vcol + 48 * N_CTX);
      z0 = wmma_f32(pa, v0, z0);
      z1 = wmma_f32(pa, v1, z1);
      z2 = wmma_f32(pa, v2, z2);
      z3 = wmma_f32(pa, v3, z3);
    }
  }
  // Store Z tile to [B, S, H*64] for the output projection
  const int b = head / N_HEAD, h = head % N_HEAD;
#pragma unroll
  for (int r = 0; r < 8; ++r) {
    const int s = qt * 16 + r + 8 * hi;
    float* zr = Z + (size_t)(b * N_CTX + s) * D_MODEL + h * D_HEAD + lo;
    zr[0]  = z0[r];
    zr[16] = z1[r];
    zr[32] = z2[r];
    zr[48] = z3[r];
  }
}

extern "C" void kernel_launch(void* const* d_in, const int* in_sizes, int n_in,
                              void* d_out, int out_size, void* d_ws, size_t ws_size,
                              hipStream_t stream) {
  const float* x  = (const float*)d_in[0];
  const float* Wq = (const float*)d_in[1];
  const float* bq = (const float*)d_in[2];
  const float* Wk = (const float*)d_in[3];
  const float* bk = (const float*)d_in[4];
  const float* Wv = (const float*)d_in[5];
  const float* bv = (const float*)d_in[6];
  const float* Wo = (const float*)d_in[7];
  const float* g  = (const float*)d_in[8];
  float* out = (float*)d_out;

  const size_t QKV = (size_t)BATCH * N_HEAD * N_CTX * D_HEAD;  // 3,145,728 floats
  float* q = (float*)d_ws;
  float* k = q + QKV;
  float* v = k + QKV;   // [B, H, D_HEAD, S] (transposed)
  float* z = v + QKV;   // [B, S, 768]

  dim3 gblk(M_ROWS / 32, D_MODEL / 256);          // (128, 3)
  qkv_gemm_kernel<<<gblk, 256, 0, stream>>>(x, Wq, bq, q, 0);
  qkv_gemm_kernel<<<gblk, 256, 0, stream>>>(x, Wk, bk, k, 0);
  qkv_gemm_kernel<<<gblk, 256, 0, stream>>>(x, Wv, bv, v, 1);

  const int rows = BATCH * N_HEAD * N_CTX;        // 49152
  norm_rope_kernel<<<rows / 8, 256, 0, stream>>>(q, g);
  norm_rope_kernel<<<rows / 8, 256, 0, stream>>>(k, g);

  attn_kernel<<<(BATCH * N_HEAD * (N_CTX / 16)) / 8, 256, 0, stream>>>(q, k, v, z);

  out_gemm_kernel<<<gblk, 256, 0, stream>>>(z, Wo, out);
}